// TransR_2602750181985
// MI455X (gfx1250) — compile-verified
//
#include <hip/hip_runtime.h>

#define NUM_RELS 1000
#define NFEATS   128
#define RFEATS   64
#define BATCH    32768

// sproj LDS layout: "slot" = a pair of K rows (slot = k>>1), 64 slots.
// Within a slot: [n][p] with p = k&1, i.e. each lane's B fragment (K pair at
// fixed N) is one contiguous 8-byte v2f -> single ds_load_b64.
// SLOT_STRIDE = 160 floats: half-wave offset (1 slot) = 160 mod 64 = 32 banks,
// so the two 16-lane halves hit disjoint bank sets -> conflict-free b64 reads.
#define SPROJ_SLOT  160
// sdiff stride 132: consecutive M rows land 4 banks apart -> conflict-free.
#define DIFF_STRIDE 132

typedef float v2f __attribute__((ext_vector_type(2)));
typedef float v8f __attribute__((ext_vector_type(8)));

// ---------------- bucketing kernels ----------------

__global__ void k_zero(int* __restrict__ p, int n) {
    int i = blockIdx.x * blockDim.x + threadIdx.x;
    if (i < n) p[i] = 0;
}

__global__ void k_hist(const long long* __restrict__ rels, int* __restrict__ counts, int n) {
    int i = blockIdx.x * blockDim.x + threadIdx.x;
    if (i < n) atomicAdd(&counts[(int)rels[i]], 1);
}

// single block, 1024 threads: exclusive scan of counts -> offsets[0..nrel], cursor copy
__global__ void k_scan(const int* __restrict__ counts, int* __restrict__ offsets,
                       int* __restrict__ cursor, int nrel) {
    __shared__ int tmp[1024];
    int t = threadIdx.x;
    int v = (t < nrel) ? counts[t] : 0;
    tmp[t] = v;
    __syncthreads();
    for (int off = 1; off < 1024; off <<= 1) {
        int x = (t >= off) ? tmp[t - off] : 0;
        __syncthreads();
        tmp[t] += x;
        __syncthreads();
    }
    int excl = tmp[t] - v;        // exclusive prefix
    if (t < nrel) { offsets[t] = excl; cursor[t] = excl; }
    if (t == nrel) offsets[nrel] = excl;   // total (v==0 here)
}

__global__ void k_scatter(const long long* __restrict__ rels, int* __restrict__ cursor,
                          int* __restrict__ perm, int n) {
    int i = blockIdx.x * blockDim.x + threadIdx.x;
    if (i < n) {
        int r = (int)rels[i];
        int pos = atomicAdd(&cursor[r], 1);
        perm[pos] = i;
    }
}

// ---------------- main WMMA kernel ----------------
// One block per relation. 4 wave32s; each wave owns a 16-row tile of the bucket.
// D[16x64] = (head-tail)[16x128] @ proj[128x64] via V_WMMA_F32_16X16X4_F32.

__global__ __launch_bounds__(128) void transr_main(
    const float* __restrict__ head, const float* __restrict__ tail,
    const float* __restrict__ relemb, const float* __restrict__ proj,
    const int* __restrict__ offsets, const int* __restrict__ perm,
    float* __restrict__ out) {

    __shared__ float sproj[64 * SPROJ_SLOT];            // 40,960 B, K-pair interleaved
    __shared__ float sdiff[4][16 * DIFF_STRIDE];        // 33,792 B

    const int rel   = blockIdx.x;
    const int start = offsets[rel];
    const int nrows = offsets[rel + 1] - start;
    if (nrows == 0) return;                              // uniform across block

    // Stage proj[rel] (128x64 f32 = 32KB) into LDS in fragment-native layout:
    // element (k, n) -> sproj[(k>>1)*SPROJ_SLOT + n*2 + (k&1)]
    {
        const float4* gp = (const float4*)(proj + (size_t)rel * (NFEATS * RFEATS));
        for (int i = threadIdx.x; i < (NFEATS * RFEATS) / 4; i += 128) {
            float4 v = gp[i];
            int k = i >> 4;            // 16 float4 per 64-wide row
            int c = (i & 15) << 2;     // n of v.x
            float* dst = &sproj[(k >> 1) * SPROJ_SLOT + (k & 1) + c * 2];
            dst[0] = v.x; dst[2] = v.y; dst[4] = v.z; dst[6] = v.w;
        }
    }
    __syncthreads();

    const int wave = threadIdx.x >> 5;
    const int lane = threadIdx.x & 31;
    const int half = lane >> 4;       // which 16-lane half
    const int lid  = lane & 15;
    float* const sd = sdiff[wave];

    // h_rel fragment: per lane, one value per 16-wide N tile (N = lid)
    float hr[4];
#pragma unroll
    for (int t = 0; t < 4; ++t) hr[t] = relemb[rel * RFEATS + t * 16 + lid];

    for (int g0 = wave * 16; g0 < nrows; g0 += 64) {
        // Stage this wave's 16 diff rows (16 x 128 f32), coalesced float4 loads.
#pragma unroll 4
        for (int r = 0; r < 16; ++r) {
            int row = g0 + r;
            int idx = perm[start + (row < nrows ? row : nrows - 1)];
            float4 h  = ((const float4*)(head + (size_t)idx * NFEATS))[lane];
            float4 t4 = ((const float4*)(tail + (size_t)idx * NFEATS))[lane];
            float* d = &sd[r * DIFF_STRIDE + lane * 4];
            d[0] = h.x - t4.x; d[1] = h.y - t4.y;
            d[2] = h.z - t4.z; d[3] = h.w - t4.w;
        }
        // wave-private LDS region; same-wave DS ordering handled by compiler waits

        v8f acc[4] = {};   // four 16x16 f32 C/D tiles covering N=0..63

#pragma unroll 4
        for (int kc = 0; kc < NFEATS / 4; ++kc) {
            // A fragment (16x4 f32): lane -> M=lid, VGPR0/1 -> K = 4kc + 2*half + {0,1}
            v2f a = *(const v2f*)&sd[lid * DIFF_STRIDE + kc * 4 + half * 2];
            // B fragment (4x16 f32): lane -> N, VGPR0/1 -> K pair; contiguous in LDS
            const float* bp = &sproj[(kc * 2 + half) * SPROJ_SLOT + lid * 2];
#pragma unroll
            for (int t = 0; t < 4; ++t) {
                v2f b = *(const v2f*)(bp + t * 32);
                acc[t] = __builtin_amdgcn_wmma_f32_16x16x4_f32(
                    /*neg_a=*/false, a, /*neg_b=*/false, b,
                    /*c_mod=*/(short)0, acc[t],
                    /*reuse_a=*/false, /*reuse_b=*/false);
            }
        }

        // Epilogue: per C/D VGPR v, lane holds D[M = v + 8*half][N = lid].
#pragma unroll
        for (int v = 0; v < 8; ++v) {
            float s = 0.0f;
#pragma unroll
            for (int t = 0; t < 4; ++t) s += fabsf(acc[t][v] + hr[t]);
            // reduce over the 16 N-lanes of each half-wave
            s += __shfl_xor(s, 1, 32);
            s += __shfl_xor(s, 2, 32);
            s += __shfl_xor(s, 4, 32);
            s += __shfl_xor(s, 8, 32);
            int row = g0 + v + half * 8;
            if (lid == 0 && row < nrows) out[perm[start + row]] = -s;
        }
    }
}

// ---------------- launch ----------------

extern "C" void kernel_launch(void* const* d_in, const int* in_sizes, int n_in,
                              void* d_out, int out_size, void* d_ws, size_t ws_size,
                              hipStream_t stream) {
    const float*     h_head  = (const float*)d_in[0];
    const float*     h_tail  = (const float*)d_in[1];
    const long long* rels    = (const long long*)d_in[2];   // jnp.int64
    const float*     rel_emb = (const float*)d_in[3];
    const float*     projtab = (const float*)d_in[4];
    float*           out     = (float*)d_out;

    int* ws      = (int*)d_ws;
    int* counts  = ws;                         // [NUM_RELS]
    int* offsets = counts + NUM_RELS;          // [NUM_RELS + 1]
    int* cursor  = offsets + NUM_RELS + 1;     // [NUM_RELS]
    int* perm    = cursor + NUM_RELS;          // [BATCH]

    k_zero<<<(NUM_RELS + 255) / 256, 256, 0, stream>>>(counts, NUM_RELS);
    k_hist<<<(BATCH + 255) / 256, 256, 0, stream>>>(rels, counts, BATCH);
    k_scan<<<1, 1024, 0, stream>>>(counts, offsets, cursor, NUM_RELS);
    k_scatter<<<(BATCH + 255) / 256, 256, 0, stream>>>(rels, cursor, perm, BATCH);
    transr_main<<<NUM_RELS, 128, 0, stream>>>(h_head, h_tail, rel_emb, projtab,
                                              offsets, perm, out);
}